// MoETransceiverVQ_54090818126069
// MI455X (gfx1250) — compile-verified
//
#include <hip/hip_runtime.h>
#include <math.h>

#define B_    128
#define N_    2048
#define INDIM 256
#define HID_  128
#define R_    8
#define MP_   4
#define K_    256
#define D_    64
#define PAD   68   // padded LDS row stride (floats): 16B-aligned float4, conflict-free b64

typedef __attribute__((ext_vector_type(2))) float v2f;
typedef __attribute__((ext_vector_type(8))) float v8f;
typedef __attribute__((ext_vector_type(4))) unsigned int u32x4;
typedef __attribute__((ext_vector_type(8))) int i32x8;
typedef __attribute__((ext_vector_type(4))) int i32x4;

__device__ __forceinline__ float gelu_exact(float x) {
    return 0.5f * x * (1.0f + erff(x * 0.70710678118654752f));
}

// ---------------- Kernel 1: router -> expert_idx[B], scale[B] ----------------
__global__ __launch_bounds__(128) void router_kernel(
    const float* __restrict__ phi, const float* __restrict__ lng, const float* __restrict__ lnb,
    const float* __restrict__ W1, const float* __restrict__ b1,
    const float* __restrict__ W2, const float* __restrict__ b2,
    const float* __restrict__ We, const float* __restrict__ be,
    const float* __restrict__ Wm, const float* __restrict__ bm,
    int* __restrict__ eidx, float* __restrict__ scal)
{
    __shared__ float s_x[INDIM];
    __shared__ float s_h1[HID_];
    __shared__ float s_h2[HID_];
    __shared__ float s_red[128];
    __shared__ float s_le[R_];
    __shared__ float s_lm[MP_];
    const int b = blockIdx.x, tid = threadIdx.x;
    float p0 = phi[b * INDIM + tid];
    float p1 = phi[b * INDIM + tid + 128];
    s_red[tid] = p0 + p1; __syncthreads();
    for (int s = 64; s > 0; s >>= 1) { if (tid < s) s_red[tid] += s_red[tid + s]; __syncthreads(); }
    float mu = s_red[0] * (1.0f / INDIM); __syncthreads();
    float d0 = p0 - mu, d1 = p1 - mu;
    s_red[tid] = d0 * d0 + d1 * d1; __syncthreads();
    for (int s = 64; s > 0; s >>= 1) { if (tid < s) s_red[tid] += s_red[tid + s]; __syncthreads(); }
    float rstd = rsqrtf(s_red[0] * (1.0f / INDIM) + 1e-5f); __syncthreads();
    s_x[tid]       = d0 * rstd * lng[tid]       + lnb[tid];
    s_x[tid + 128] = d1 * rstd * lng[tid + 128] + lnb[tid + 128];
    __syncthreads();
    float acc = b1[tid];
    for (int i = 0; i < INDIM; ++i) acc += s_x[i] * W1[i * HID_ + tid];
    s_h1[tid] = gelu_exact(acc);
    __syncthreads();
    acc = b2[tid];
    for (int i = 0; i < HID_; ++i) acc += s_h1[i] * W2[i * HID_ + tid];
    s_h2[tid] = gelu_exact(acc);
    __syncthreads();
    if (tid < R_) {
        float a = be[tid];
        for (int i = 0; i < HID_; ++i) a += s_h2[i] * We[i * R_ + tid];
        s_le[tid] = a;
    } else if (tid < R_ + MP_) {
        int m = tid - R_;
        float a = bm[m];
        for (int i = 0; i < HID_; ++i) a += s_h2[i] * Wm[i * MP_ + m];
        s_lm[m] = a;
    }
    __syncthreads();
    if (tid == 0) {
        int best = 0; float bv = s_le[0];
        for (int r = 1; r < R_; ++r) if (s_le[r] > bv) { bv = s_le[r]; best = r; }
        eidx[b] = best;
        float mx = s_lm[0];
        for (int m = 1; m < MP_; ++m) mx = fmaxf(mx, s_lm[m]);
        float Z = 0.f, e[MP_];
        for (int m = 0; m < MP_; ++m) { e[m] = __expf(s_lm[m] - mx); Z += e[m]; }
        const float POWER[MP_] = {0.25f, 0.5f, 0.75f, 1.0f};
        float sc = 0.f;
        for (int m = 0; m < MP_; ++m) sc += (e[m] / Z) * POWER[m];
        scal[b] = sc;
    }
}

// ------- Kernel 2: fused VQ-WMMA + softmax + soft-bits + QAM symbols --------
// grid = B * (N/128) blocks, 256 threads (8 waves); wave w owns 16 rows of n.
// Expert codebook staged into LDS by the Tensor Data Mover with hardware row
// padding (interval 64 DWORDs, pad 4 DWORDs -> stride 68 floats, conflict-free).
__global__ __launch_bounds__(256) void vq_mod_kernel(
    const float* __restrict__ z, const float* __restrict__ codebooks,
    const int* __restrict__ eidx, const float* __restrict__ scal,
    float* __restrict__ out)
{
    extern __shared__ float smem[];
    float* s_cb  = smem;                 // [K_][PAD]
    float* s_cb2 = s_cb + K_ * PAD;      // [K_]
    float* s_z   = s_cb2 + K_;           // [8][16][PAD]

    const int blk_per_b = N_ / 128;      // 16
    const int b    = blockIdx.x / blk_per_b;
    const int nblk = blockIdx.x % blk_per_b;
    const int tid  = threadIdx.x;
    const int w    = tid >> 5;
    const int lane = tid & 31;
    const int half = lane >> 4;
    const int lm   = lane & 15;

    // ---- stage expert codebook via TDM (wave 0 only; TDM issues once/wave) ----
    if (w == 0) {
        unsigned e = (unsigned)__builtin_amdgcn_readfirstlane(eidx[b]);
        unsigned long long gaddr =
            (unsigned long long)(size_t)codebooks + (unsigned long long)e * (K_ * D_ * 4ull);
        unsigned lds_off = (unsigned)(size_t)s_cb;   // low 32 bits of generic = LDS byte offset
        // D# group0: count=1 | lds_addr | global_addr[56:0] | type=2
        u32x4 g0 = { 1u, lds_off, (unsigned)gaddr,
                     (unsigned)((gaddr >> 32) & 0x01FFFFFFull) | (2u << 30) };
        // D# group1: data_size=4B(code2), pad_enable, pad_interval=64dw(code5),
        //            pad_amount=4dw(code3); tensor 64x256, tile 64x256, stride0=64
        i32x8 g1 = { (int)((2u << 16) | (1u << 20) | (5u << 22) | (3u << 25)),
                     (int)(D_ << 16),        // tensor_dim0 = 64  (bits 79:48)
                     (int)(K_ << 16),        // tensor_dim1 = 256 (bits 111:80)
                     (int)(D_ << 16),        // tile_dim0   = 64  (bits 127:112)
                     (int)K_,                // tile_dim1   = 256 (bits 143:128)
                     (int)D_,                // tensor_dim0_stride = 64 (bits 207:160)
                     0, 0 };
        i32x4 g2 = { 0, 0, 0, 0 };
        i32x4 g3 = { 0, 0, 0, 0 };
        i32x8 g4 = { 0, 0, 0, 0, 0, 0, 0, 0 };
        __builtin_amdgcn_tensor_load_to_lds(g0, g1, g2, g3, g4, 0);
    }
    // ---- stage z slab (overlaps TDM): wave w -> rows 16w..16w+15 ----
    const float* zg = z + ((size_t)b * N_ + (size_t)nblk * 128 + (size_t)w * 16) * D_;
    float* s_zw = s_z + w * 16 * PAD;
    for (int it = 0; it < 8; ++it) {
        int flat = it * 128 + lane * 4;
        int row = flat >> 6, col = flat & 63;
        float4 v = *(const float4*)(zg + flat);
        *(float4*)&s_zw[row * PAD + col] = v;
    }
    if (w == 0) __builtin_amdgcn_s_wait_tensorcnt(0);
    __syncthreads();
    // ---- ||cb_k||^2 ----
    {
        float s = 0.f;
        const float* rp = &s_cb[tid * PAD];
        for (int d0 = 0; d0 < D_; ++d0) s += rp[d0] * rp[d0];
        s_cb2[tid] = s;
    }
    __syncthreads();

    // ---- WMMA: 16x256 scores = z_tile(16x64) @ cb^T(64x256) ----
    v8f acc[16];
    #pragma unroll
    for (int t = 0; t < 16; ++t) acc[t] = (v8f){0.f,0.f,0.f,0.f,0.f,0.f,0.f,0.f};

    for (int kk = 0; kk < 16; ++kk) {                    // K-dim: 4 per step
        v2f a = *(const v2f*)&s_zw[lm * PAD + 4 * kk + 2 * half];
        v2f bb[16];
        #pragma unroll
        for (int t = 0; t < 16; ++t)                     // batch all B loads first
            bb[t] = *(const v2f*)&s_cb[(t * 16 + lm) * PAD + 4 * kk + 2 * half];
        #pragma unroll
        for (int t = 0; t < 16; ++t)                     // then the WMMA burst
            acc[t] = __builtin_amdgcn_wmma_f32_16x16x4_f32(
                false, a, false, bb[t], (short)0, acc[t], false, false);
    }

    float ccol[16];
    #pragma unroll
    for (int t = 0; t < 16; ++t) ccol[t] = s_cb2[t * 16 + lm];

    const float sc   = scal[b];
    const float invs = 0.23570226039551584f;             // 1/sqrt(18)
    float cbit[4];
    #pragma unroll
    for (int j = 0; j < 4; ++j) cbit[j] = (float)((lm >> (3 - j)) & 1);
    const float candx = (2.0f * (float)(lm >> 2) - 3.0f) * invs;
    const float candy = (2.0f * (float)(lm & 3)  - 3.0f) * invs;
    const float pc2   = 2.0f * (float)__popc(lm);
    const int n_row_base = nblk * 128 + w * 16;

    #pragma unroll
    for (int vg = 0; vg < 8; ++vg) {                     // rows vg (lanes<16), vg+8 (lanes>=16)
        float v[16];
        #pragma unroll
        for (int t = 0; t < 16; ++t) v[t] = 4.0f * acc[t][vg] - 2.0f * ccol[t];
        float mx = v[0];
        #pragma unroll
        for (int t = 1; t < 16; ++t) mx = fmaxf(mx, v[t]);
        #pragma unroll
        for (int m = 8; m >= 1; m >>= 1) mx = fmaxf(mx, __shfl_xor(mx, m, 32));
        float E = 0.f, tb0 = 0.f, tb1 = 0.f, tb2 = 0.f, tb3 = 0.f;
        #pragma unroll
        for (int t = 0; t < 16; ++t) {
            float ev = __expf(v[t] - mx);
            E += ev;
            if (t & 8) tb0 += ev;   // label bit c=0 (MSB of k) == bit3 of t
            if (t & 4) tb1 += ev;
            if (t & 2) tb2 += ev;
            if (t & 1) tb3 += ev;
        }
        float lb0 = cbit[0] * E, lb1 = cbit[1] * E, lb2 = cbit[2] * E, lb3 = cbit[3] * E;
        #pragma unroll
        for (int m = 8; m >= 1; m >>= 1) {
            E   += __shfl_xor(E,   m, 32);
            tb0 += __shfl_xor(tb0, m, 32);  tb1 += __shfl_xor(tb1, m, 32);
            tb2 += __shfl_xor(tb2, m, 32);  tb3 += __shfl_xor(tb3, m, 32);
            lb0 += __shfl_xor(lb0, m, 32);  lb1 += __shfl_xor(lb1, m, 32);
            lb2 += __shfl_xor(lb2, m, 32);  lb3 += __shfl_xor(lb3, m, 32);
        }
        const float invZ = 1.0f / E;
        const float sb0 = tb0 * invZ, sb1 = tb1 * invZ, sb2 = tb2 * invZ, sb3 = tb3 * invZ;
        const float sb4 = lb0 * invZ, sb5 = lb1 * invZ, sb6 = lb2 * invZ, sb7 = lb3 * invZ;
        float g0 = 4.0f * (sb0 * cbit[0] + sb1 * cbit[1] + sb2 * cbit[2] + sb3 * cbit[3]) - pc2;
        float g1 = 4.0f * (sb4 * cbit[0] + sb5 * cbit[1] + sb6 * cbit[2] + sb7 * cbit[3]) - pc2;
        float e0 = __expf(g0), e1 = __expf(g1);
        float X0 = e0 * candx, Y0 = e0 * candy, Z0 = e0;
        float X1 = e1 * candx, Y1 = e1 * candy, Z1 = e1;
        #pragma unroll
        for (int m = 8; m >= 1; m >>= 1) {
            X0 += __shfl_xor(X0, m, 32); Y0 += __shfl_xor(Y0, m, 32); Z0 += __shfl_xor(Z0, m, 32);
            X1 += __shfl_xor(X1, m, 32); Y1 += __shfl_xor(Y1, m, 32); Z1 += __shfl_xor(Z1, m, 32);
        }
        if (lm == 0) {
            int n = n_row_base + vg + 8 * half;
            float4 o;
            o.x = X0 / Z0 * sc;  o.y = Y0 / Z0 * sc;   // symbol 2n
            o.z = X1 / Z1 * sc;  o.w = Y1 / Z1 * sc;   // symbol 2n+1
            *(float4*)(out + ((size_t)b * 4096 + 2 * (size_t)n) * 2) = o;
        }
    }
}

extern "C" void kernel_launch(void* const* d_in, const int* in_sizes, int n_in,
                              void* d_out, int out_size, void* d_ws, size_t ws_size,
                              hipStream_t stream) {
    (void)in_sizes; (void)n_in; (void)out_size; (void)ws_size;
    const float* phi = (const float*)d_in[0];
    const float* z   = (const float*)d_in[1];
    const float* lng = (const float*)d_in[2];
    const float* lnb = (const float*)d_in[3];
    const float* W1  = (const float*)d_in[4];
    const float* b1  = (const float*)d_in[5];
    const float* W2  = (const float*)d_in[6];
    const float* b2  = (const float*)d_in[7];
    const float* We  = (const float*)d_in[8];
    const float* be  = (const float*)d_in[9];
    const float* Wm  = (const float*)d_in[10];
    const float* bm  = (const float*)d_in[11];
    const float* cbs = (const float*)d_in[12];

    int*   eidx = (int*)d_ws;
    float* scal = (float*)((char*)d_ws + B_ * sizeof(int));
    float* out  = (float*)d_out;

    router_kernel<<<B_, 128, 0, stream>>>(phi, lng, lnb, W1, b1, W2, b2,
                                          We, be, Wm, bm, eidx, scal);

    size_t shmem = (size_t)(K_ * PAD + K_ + 8 * 16 * PAD) * sizeof(float); // ~103 KB (WGP has 320 KB)
    vq_mod_kernel<<<B_ * (N_ / 128), 256, shmem, stream>>>(z, cbs, eidx, scal, out);
}